// Encoder_24842090840532
// MI455X (gfx1250) — compile-verified
//
#include <hip/hip_runtime.h>
#include <math.h>

typedef __attribute__((ext_vector_type(2))) float v2f;
typedef __attribute__((ext_vector_type(8))) float v8f;

// ---------------------------------------------------------------------------
// degree / norm helpers
// ---------------------------------------------------------------------------
__global__ void k_fill(float* __restrict__ p, float v, long long n) {
    long long i = (long long)blockIdx.x * blockDim.x + threadIdx.x;
    if (i < n) p[i] = v;
}

__global__ void k_degree(const int* __restrict__ dst, float* __restrict__ deg, int E) {
    int e = blockIdx.x * blockDim.x + threadIdx.x;
    if (e < E) atomicAdd(&deg[dst[e]], 1.0f);
}

__global__ void k_dinv(const float* __restrict__ deg, float* __restrict__ dinv, int n) {
    int i = blockIdx.x * blockDim.x + threadIdx.x;
    if (i < n) {
        float d = deg[i];
        dinv[i] = (d > 0.0f) ? rsqrtf(d) : 0.0f;
    }
}

// ---------------------------------------------------------------------------
// C[M x Nc] = A[M x K] @ B[K x Nc], fp32, via V_WMMA_F32_16X16X4_F32.
// One wave (32 lanes) computes one 16x16 output tile. M,K,Nc multiples of 16.
// blockDim = (32, WAVES_PER_BLOCK). Tile id derived from threadIdx.y only so
// the early-out is wave-uniform (EXEC all-ones at every WMMA).
// ---------------------------------------------------------------------------
#define WAVES_PER_BLOCK 4

__global__ void k_gemm_wmma_f32(const float* __restrict__ A,
                                const float* __restrict__ B,
                                float* __restrict__ C,
                                int M, int K, int Nc) {
    const int lane   = threadIdx.x & 31;
    const int tilesN = Nc >> 4;
    const int tilesM = M >> 4;
    const int tile   = blockIdx.x * WAVES_PER_BLOCK + threadIdx.y;
    if (tile >= tilesM * tilesN) return;          // wave-uniform exit

    const int tm  = tile / tilesN;
    const int tn  = tile - tm * tilesN;
    const int grp = lane >> 4;                    // 0: lanes 0-15, 1: lanes 16-31
    const int hl  = lane & 15;

    const float* arow = A + (size_t)(tm * 16 + hl) * K;   // A row for this lane
    const int n = tn * 16 + hl;                            // B/C column for this lane

    v8f acc = {};
    for (int k0 = 0; k0 < K; k0 += 4) {
        const int ka = k0 + 2 * grp;                       // K index pair for this half-wave
        v2f a = *(const v2f*)(arow + ka);                  // A[m][ka], A[m][ka+1]
        v2f b;
        b.x = B[(size_t)ka * Nc + n];                      // B[ka][n]
        b.y = B[(size_t)(ka + 1) * Nc + n];                // B[ka+1][n]
        acc = __builtin_amdgcn_wmma_f32_16x16x4_f32(
            /*neg_a=*/false, a, /*neg_b=*/false, b,
            /*c_mod=*/(short)0, acc, /*reuse_a=*/false, /*reuse_b=*/false);
    }

    // C/D layout: VGPR r, lanes 0-15 -> row tm*16+r ; lanes 16-31 -> row tm*16+8+r
    float* crow = C + (size_t)(tm * 16 + 8 * grp) * Nc + n;
#pragma unroll
    for (int r = 0; r < 8; ++r) crow[(size_t)r * Nc] = acc[r];
}

// ---------------------------------------------------------------------------
// Edge scatter: acc[dst] += dinv[src]*dinv[dst] * h[src], one thread per
// (edge, feature). F is a power of two (64 or 128) -> logF shifts.
// ---------------------------------------------------------------------------
__global__ void k_scatter(const float* __restrict__ h,
                          const int* __restrict__ src,
                          const int* __restrict__ dst,
                          const float* __restrict__ dinv,
                          float* __restrict__ acc,
                          long long total, int F, int logF) {
    long long idx = (long long)blockIdx.x * blockDim.x + threadIdx.x;
    if (idx >= total) return;
    int e = (int)(idx >> logF);
    int f = (int)(idx & (F - 1));
    int s = src[e], d = dst[e];
    float w = dinv[s] * dinv[d];
    atomicAdd(&acc[(size_t)d * F + f], w * h[(size_t)s * F + f]);
}

// Self-loop contribution (norm = dinv[i]^2) + bias + optional relu.
// out may alias acc.
__global__ void k_finalize(const float* __restrict__ acc,
                           const float* __restrict__ h,
                           const float* __restrict__ dinv,
                           const float* __restrict__ bias,
                           float* __restrict__ out,
                           long long total, int F, int logF, int relu) {
    long long idx = (long long)blockIdx.x * blockDim.x + threadIdx.x;
    if (idx >= total) return;
    int i = (int)(idx >> logF);
    int f = (int)(idx & (F - 1));
    float di = dinv[i];
    float v = acc[idx] + di * di * h[idx] + bias[f];
    if (relu) v = fmaxf(v, 0.0f);
    out[idx] = v;
}

// logvar = tanh(lpre + bl); z = noise * exp(0.5*logvar) + mean
__global__ void k_reparam(const float* __restrict__ lpre,
                          const float* __restrict__ bl,
                          const float* __restrict__ noise,
                          const float* __restrict__ mean,
                          float* __restrict__ logvar_out,
                          float* __restrict__ z_out,
                          long long total, int H, int logH) {
    long long idx = (long long)blockIdx.x * blockDim.x + threadIdx.x;
    if (idx >= total) return;
    int f = (int)(idx & (H - 1));
    float lv = tanhf(lpre[idx] + bl[f]);
    logvar_out[idx] = lv;
    z_out[idx] = noise[idx] * expf(0.5f * lv) + mean[idx];
}

// ---------------------------------------------------------------------------
// Host launch
// ---------------------------------------------------------------------------
static inline dim3 grid1d(long long total, int block) {
    return dim3((unsigned)((total + block - 1) / block));
}

extern "C" void kernel_launch(void* const* d_in, const int* in_sizes, int n_in,
                              void* d_out, int out_size, void* d_ws, size_t ws_size,
                              hipStream_t stream) {
    const float* x     = (const float*)d_in[0];
    const int*   eidx  = (const int*)  d_in[1];
    const float* noise = (const float*)d_in[2];
    const float* W1    = (const float*)d_in[3];
    const float* b1    = (const float*)d_in[4];
    const float* Wm    = (const float*)d_in[5];
    const float* bm    = (const float*)d_in[6];
    const float* Wv    = (const float*)d_in[7];
    const float* bv    = (const float*)d_in[8];
    const float* Wl    = (const float*)d_in[9];
    const float* bl    = (const float*)d_in[10];

    const int H  = in_sizes[6];            // 64
    const int F2 = in_sizes[4];            // 2H = 128
    const int Fd = in_sizes[3] / F2;       // 128
    const int N  = in_sizes[0] / Fd;       // 50000
    const int E  = in_sizes[1] / 2;        // 800000
    const int*  src = eidx;
    const int*  dst = eidx + E;

    const int logFd = 7, logH2 = 7, logH = 6;  // Fd=128, 2H=128, H=64

    // workspace layout (floats)
    float* ws    = (float*)d_ws;
    float* deg   = ws;                               // N
    float* dinv  = deg  + N;                         // N
    float* h1    = dinv + N;                         // N*F2   (x@W1)
    float* acc1  = h1   + (size_t)N * F2;            // N*F2   (-> hidden, in place)
    float* hm    = acc1 + (size_t)N * F2;            // N*H    (hidden@Wm; reused as lpre)
    float* hv    = hm   + (size_t)N * H;             // N*H    (hidden@Wv)
    float* accm  = hv   + (size_t)N * H;             // N*H
    float* accv  = accm + (size_t)N * H;             // N*H    (-> relu_v, in place)

    float* mean_out   = (float*)d_out;               // N*H
    float* logvar_out = mean_out + (size_t)N * H;    // N*H
    float* z_out      = logvar_out + (size_t)N * H;  // N*H

    const int B = 256;
    const dim3 wblock(32, WAVES_PER_BLOCK);

    // 1) degree with self-loop (start at 1), then dinv = rsqrt(deg)
    k_fill<<<grid1d(N, B), B, 0, stream>>>(deg, 1.0f, N);
    k_degree<<<grid1d(E, B), B, 0, stream>>>(dst, deg, E);
    k_dinv<<<grid1d(N, B), B, 0, stream>>>(deg, dinv, N);

    // 2) layer 1: h1 = x @ W1 ; aggregate ; hidden = relu(agg + self + b1)
    {
        long long tiles = (long long)(N / 16) * (F2 / 16);
        k_gemm_wmma_f32<<<grid1d(tiles, WAVES_PER_BLOCK), wblock, 0, stream>>>(
            x, W1, h1, N, Fd, F2);
        long long tot = (long long)N * F2;
        k_fill<<<grid1d(tot, B), B, 0, stream>>>(acc1, 0.0f, tot);
        long long work = (long long)E * F2;
        k_scatter<<<grid1d(work, B), B, 0, stream>>>(h1, src, dst, dinv, acc1,
                                                     work, F2, logH2);
        k_finalize<<<grid1d(tot, B), B, 0, stream>>>(acc1, h1, dinv, b1, acc1,
                                                     tot, F2, logH2, /*relu=*/1);
    }
    float* hidden = acc1;

    // 3) mean & logvar branches: hm = hidden@Wm, hv = hidden@Wv
    {
        long long tiles = (long long)(N / 16) * (H / 16);
        k_gemm_wmma_f32<<<grid1d(tiles, WAVES_PER_BLOCK), wblock, 0, stream>>>(
            hidden, Wm, hm, N, F2, H);
        k_gemm_wmma_f32<<<grid1d(tiles, WAVES_PER_BLOCK), wblock, 0, stream>>>(
            hidden, Wv, hv, N, F2, H);

        long long tot  = (long long)N * H;
        long long work = (long long)E * H;
        k_fill<<<grid1d(tot, B), B, 0, stream>>>(accm, 0.0f, tot);
        k_fill<<<grid1d(tot, B), B, 0, stream>>>(accv, 0.0f, tot);
        k_scatter<<<grid1d(work, B), B, 0, stream>>>(hm, src, dst, dinv, accm,
                                                     work, H, logH);
        k_scatter<<<grid1d(work, B), B, 0, stream>>>(hv, src, dst, dinv, accv,
                                                     work, H, logH);
        // mean -> d_out directly (no relu); relu_v -> accv in place
        k_finalize<<<grid1d(tot, B), B, 0, stream>>>(accm, hm, dinv, bm, mean_out,
                                                     tot, H, logH, /*relu=*/0);
        k_finalize<<<grid1d(tot, B), B, 0, stream>>>(accv, hv, dinv, bv, accv,
                                                     tot, H, logH, /*relu=*/1);
    }

    // 4) lpre = relu_v @ Wl (reuse hm buffer), then tanh/exp/reparameterize
    {
        long long tiles = (long long)(N / 16) * (H / 16);
        k_gemm_wmma_f32<<<grid1d(tiles, WAVES_PER_BLOCK), wblock, 0, stream>>>(
            accv, Wl, hm, N, H, H);
        long long tot = (long long)N * H;
        k_reparam<<<grid1d(tot, B), B, 0, stream>>>(hm, bl, noise, mean_out,
                                                    logvar_out, z_out, tot, H, logH);
    }
}